// GumbelSoftmaxPaletteQuantization_15272903704771
// MI455X (gfx1250) — compile-verified
//
#include <hip/hip_runtime.h>

#define GUMBEL_EPS 1e-20f

typedef float v2f __attribute__((ext_vector_type(2)));
typedef float v8f __attribute__((ext_vector_type(8)));

// b=8, h=256, w=256, k=256, c=4
#define KDIM 256
#define CDIM 4
#define NPIX (256 * 256)
#define PIX_PER_BLOCK 16
#define PSTRIDE (KDIM + 4) // pad LDS rows: 260 % 64 == 4 -> spreads banks across pixel rows

__global__ __launch_bounds__(256)
void GumbelSoftmaxPaletteQuantization_kernel(const float* __restrict__ images,
                                             const float* __restrict__ palettes,
                                             const float* __restrict__ noise,
                                             const float* __restrict__ tempPtr,
                                             float* __restrict__ out)
{
    __shared__ __align__(16) float sPal[KDIM * CDIM];          // 4 KB palette for this batch
    __shared__ __align__(16) float sProb[PIX_PER_BLOCK * PSTRIDE]; // 16 pixel prob rows

    const int tid   = threadIdx.x;
    const int wave  = tid >> 5;
    const int lane  = tid & 31;
    const int batch = blockIdx.y;
    const int pixBase = blockIdx.x * PIX_PER_BLOCK;

    // ---- palette -> LDS (one float4 per k, 256 threads cover k=0..255) ----
    {
        const float4* src = (const float4*)(palettes + (size_t)batch * KDIM * CDIM);
        ((float4*)sPal)[tid] = src[tid];
    }

    const float invT = 1.0f / tempPtr[0];

    __syncthreads();

    // ---- softmax of (logits + gumbel)/T : one wave per pixel, lane owns 8 k ----
#pragma unroll
    for (int rep = 0; rep < 2; ++rep) {
        const int pix = wave * 2 + rep;                       // 0..15
        const size_t gp = (size_t)batch * NPIX + pixBase + pix;
        const float4* img4 = (const float4*)(images + gp * KDIM);
        const float4* noi4 = (const float4*)(noise  + gp * KDIM);
        const int kv = lane * 2;                              // float4 index (8 floats/lane)

        float4 i0 = img4[kv],     i1 = img4[kv + 1];
        float4 u0 = noi4[kv],     u1 = noi4[kv + 1];

        float x[8] = { i0.x, i0.y, i0.z, i0.w, i1.x, i1.y, i1.z, i1.w };
        float u[8] = { u0.x, u0.y, u0.z, u0.w, u1.x, u1.y, u1.z, u1.w };

        float l[8];
        float mx = -INFINITY;
#pragma unroll
        for (int j = 0; j < 8; ++j) {
            float g = -__logf(-__logf(u[j] + GUMBEL_EPS) + GUMBEL_EPS);
            l[j] = (x[j] + g) * invT;
            mx = fmaxf(mx, l[j]);
        }
        // wave32 max-reduce
#pragma unroll
        for (int off = 16; off > 0; off >>= 1)
            mx = fmaxf(mx, __shfl_xor(mx, off, 32));

        float s = 0.0f;
#pragma unroll
        for (int j = 0; j < 8; ++j) {
            float e = __expf(l[j] - mx);
            l[j] = e;
            s += e;
        }
        // wave32 sum-reduce
#pragma unroll
        for (int off = 16; off > 0; off >>= 1)
            s += __shfl_xor(s, off, 32);

        const float invS = 1.0f / s;
        float4 p0 = { l[0] * invS, l[1] * invS, l[2] * invS, l[3] * invS };
        float4 p1 = { l[4] * invS, l[5] * invS, l[6] * invS, l[7] * invS };

        float4* dst = (float4*)&sProb[pix * PSTRIDE + lane * 8]; // 16B aligned (1040*pix + 32*lane)
        dst[0] = p0;
        dst[1] = p1;
    }

    __syncthreads();

    // ---- contraction probs[16x256] x palette[256x4] via V_WMMA_F32_16X16X4_F32 ----
    // A (16x4 f32): lane = (k>=2 ? 16:0) + m, VGPR = k&1
    // B (4x16 f32): lane = (k>=2 ? 16:0) + n, VGPR = k&1 (only n<4 nonzero)
    // D (16x16):    lane = (m>=8 ? 16:0) + n, VGPR = m&7
    if (wave == 0) {
        const int m      = lane & 15;           // A row = pixel
        const int n      = lane & 15;           // B/D column = channel (only n<4 used)
        const int kkBase = (lane >> 4) * 2;     // 0 for lanes 0-15, 2 for lanes 16-31
        const float bMask = (n < CDIM) ? 1.0f : 0.0f;
        const int   nC    = n & (CDIM - 1);     // clamped channel index for masked loads

        v8f acc = {};
#pragma unroll 4
        for (int kb = 0; kb < KDIM; kb += 4) {
            v2f a, b;
            a.x = sProb[m * PSTRIDE + kb + kkBase];
            a.y = sProb[m * PSTRIDE + kb + kkBase + 1];
            b.x = bMask * sPal[(kb + kkBase)     * CDIM + nC];
            b.y = bMask * sPal[(kb + kkBase + 1) * CDIM + nC];
            acc = __builtin_amdgcn_wmma_f32_16x16x4_f32(
                /*neg_a=*/false, a, /*neg_b=*/false, b,
                /*c_mod=*/(short)0, acc, /*reuse_a=*/false, /*reuse_b=*/false);
        }

        // scatter D: VGPR v -> pixel (mBase+v), lane%16 -> channel
        if (n < CDIM) {
            const int mBase = (lane >> 4) * 8;  // 0 or 8
            const size_t base = ((size_t)batch * NPIX + pixBase + mBase) * CDIM + n;
#pragma unroll
            for (int v = 0; v < 8; ++v)
                out[base + (size_t)v * CDIM] = acc[v];
        }
    }
}

extern "C" void kernel_launch(void* const* d_in, const int* in_sizes, int n_in,
                              void* d_out, int out_size, void* d_ws, size_t ws_size,
                              hipStream_t stream) {
    const float* images   = (const float*)d_in[0];
    const float* palettes = (const float*)d_in[1];
    const float* noise    = (const float*)d_in[2];
    const float* temp     = (const float*)d_in[3];
    float* out = (float*)d_out;

    dim3 grid(NPIX / PIX_PER_BLOCK, 8); // 4096 x 8 blocks
    GumbelSoftmaxPaletteQuantization_kernel<<<grid, 256, 0, stream>>>(
        images, palettes, noise, temp, out);
}